// AMSTG_78924319031339
// MI455X (gfx1250) — compile-verified
//
#include <hip/hip_runtime.h>
#include <math.h>

typedef float v2f __attribute__((ext_vector_type(2)));
typedef float v8f __attribute__((ext_vector_type(8)));

#define BB 8
#define HH 48
#define NN 96
#define AA 6
#define DD 64
#define NHEAD 4
#define HID_ 128
#define DFF_ 128
#define ROWS (BB*HH*NN*AA)   /* 221184 */
#define BHDIM (BB*HH)        /* 384 */

#define XL 68     /* stride for 64-col LDS tiles (4 mod 64)   */
#define SL 52     /* stride for 48-col score tiles (52 mod 64)*/
#define FL 132    /* stride for 128-col LDS tiles (4 mod 64)  */
#define SLD 100   /* stride for 96-col adjacency (36 mod 64)  */
#define ELD 54    /* stride for 52-col feat tile (54 mod 64)  */

// ---------------- WMMA tile helpers (fp32, 16x16, K step 4) ----------------
// A: 16xK row-major (lda), B: Kx16 row-major (ldb). One wave per tile.
template<int K>
static __device__ __forceinline__ v8f wmma_acc(const float* A, int lda,
                                               const float* B, int ldb,
                                               v8f acc, int lane) {
  const int m = lane & 15;
  const int sel = lane >> 4;
#pragma unroll
  for (int kk = 0; kk < K; kk += 4) {
    v2f a, b;
    const float* ap = A + m * lda + kk + 2 * sel;
    a.x = ap[0];
    a.y = ap[1];
    const float* bp = B + (kk + 2 * sel) * ldb + m;
    b.x = bp[0];
    b.y = bp[ldb];
    acc = __builtin_amdgcn_wmma_f32_16x16x4_f32(false, a, false, b,
                                                (short)0, acc, false, false);
  }
  return acc;
}

// Same, but B given transposed: Bt is 16xK row-major (B[k][n] = Bt[n][k]).
template<int K>
static __device__ __forceinline__ v8f wmma_acc_bt(const float* A, int lda,
                                                  const float* Bt, int ldbt,
                                                  v8f acc, int lane) {
  const int m = lane & 15;
  const int sel = lane >> 4;
#pragma unroll
  for (int kk = 0; kk < K; kk += 4) {
    v2f a, b;
    const float* ap = A + m * lda + kk + 2 * sel;
    a.x = ap[0];
    a.y = ap[1];
    const float* bp = Bt + m * ldbt + kk + 2 * sel;
    b.x = bp[0];
    b.y = bp[1];
    acc = __builtin_amdgcn_wmma_f32_16x16x4_f32(false, a, false, b,
                                                (short)0, acc, false, false);
  }
  return acc;
}

static __device__ __forceinline__ void store_tile(float* D, int ldd, v8f acc, int lane) {
  const int n = lane & 15;
  const int sel = lane >> 4;
#pragma unroll
  for (int v = 0; v < 8; ++v)
    D[(v + 8 * sel) * ldd + n] = acc[v];
}

static __device__ __forceinline__ v8f bias_acc(const float* bias, int n0, int lane) {
  float bv = bias[n0 + (lane & 15)];
  v8f acc;
#pragma unroll
  for (int v = 0; v < 8; ++v) acc[v] = bv;
  return acc;
}

static __device__ __forceinline__ v8f relu8(v8f a) {
#pragma unroll
  for (int v = 0; v < 8; ++v) a[v] = fmaxf(a[v], 0.f);
  return a;
}

// ---------------- static adjacency prep ----------------
__global__ void __launch_bounds__(256) cc_attr_kernel(const float* C, float* ccA) {
  int idx = blockIdx.x * 256 + threadIdx.x;
  if (idx >= NN * AA * AA) return;
  int n = idx / 36, r = idx % 36, i = r / 6, j = r % 6;
  const float* ci = C + (n * 6 + i) * 16;
  const float* cj = C + (n * 6 + j) * 16;
  float s = 0.f;
#pragma unroll
  for (int f = 0; f < 16; ++f) s += ci[f] * cj[f];
  ccA[idx] = fmaxf(s, 0.f);
}

__global__ void __launch_bounds__(256) cc_spa_kernel(const float* C, float* ccS) {
  int idx = blockIdx.x * 256 + threadIdx.x;
  if (idx >= AA * NN * NN) return;
  int a = idx / (NN * NN), r = idx % (NN * NN), n = r / NN, m = r % NN;
  const float* cn = C + (n * 6 + a) * 16;
  const float* cm = C + (m * 6 + a) * 16;
  float s = 0.f;
#pragma unroll
  for (int f = 0; f < 16; ++f) s += cn[f] * cm[f];
  ccS[idx] = fmaxf(s, 0.f);
}

// ---------------- embedding MLP: feat(50) -> 128 -> 64 ----------------
__global__ void __launch_bounds__(256) embed_kernel(
    const float* X_in, const float* M_in, const float* U_in, const float* W_in,
    const float* C, const float* w1, const float* b1, const float* w2,
    const float* b2, float* out) {
  __shared__ float sF[64 * ELD];
  __shared__ float sW1[52 * HID_];
  __shared__ float sH1[64 * FL];
  const int tid = threadIdx.x, lane = tid & 31, w = tid >> 5;
  const size_t row0 = (size_t)blockIdx.x * 64;

  for (int i = tid; i < 52 * HID_; i += 256)
    sW1[i] = (i < 50 * HID_) ? w1[i] : 0.f;
  for (int i = tid; i < 64 * 52; i += 256) {
    int r = i / 52, c = i % 52;
    size_t row = row0 + r;
    int bh = (int)(row / (NN * AA));
    int na = (int)(row % (NN * AA));
    float v;
    if (c == 0) v = X_in[row];
    else if (c == 1) v = M_in[row];
    else if (c < 18) v = C[na * 16 + (c - 2)];
    else if (c < 34) v = U_in[bh * 16 + (c - 18)];
    else if (c < 50) v = W_in[bh * 16 + (c - 34)];
    else v = 0.f;
    sF[r * ELD + c] = v;
  }
  __syncthreads();
  for (int t = w; t < 32; t += 8) {   // 4x8 tiles, K=52
    int rt = t >> 3, ct = t & 7;
    v8f acc = bias_acc(b1, ct * 16, lane);
    acc = wmma_acc<52>(sF + rt * 16 * ELD, ELD, sW1 + ct * 16, HID_, acc, lane);
    store_tile(sH1 + rt * 16 * FL + ct * 16, FL, relu8(acc), lane);
  }
  __syncthreads();
  for (int t = w; t < 16; t += 8) {   // 4x4 tiles, K=128
    int rt = t >> 2, ct = t & 3;
    v8f acc = bias_acc(b2, ct * 16, lane);
    acc = wmma_acc<HID_>(sH1 + rt * 16 * FL, FL, w2 + ct * 16, DD, acc, lane);
    const int n = lane & 15, sel = lane >> 4;
#pragma unroll
    for (int v = 0; v < 8; ++v)
      out[(row0 + rt * 16 + v + 8 * sel) * DD + ct * 16 + n] = acc[v];
  }
}

// ---------------- temporal transformer layer (one block per (b,n,a)) -------
__global__ void __launch_bounds__(256) temporal_kernel(
    float* Hcur,
    const float* wq, const float* bq, const float* wk, const float* bk,
    const float* wv, const float* bv, const float* wo, const float* bo,
    const float* ln1g, const float* ln1b, const float* ln2g, const float* ln2b,
    const float* ff1w, const float* ff1b, const float* ff2w, const float* ff2b) {
  __shared__ float sX[48 * XL];
  __shared__ float sQ[48 * XL];
  __shared__ float sK[48 * XL];
  __shared__ float sV[48 * XL];
  __shared__ float sO[48 * XL];
  __shared__ float sP[48 * XL];
  __shared__ float sS[12][16 * SL];
  __shared__ float sH1[48 * FL];
  const int tid = threadIdx.x, lane = tid & 31, w = tid >> 5;
  const int seq = blockIdx.x;                 // (b*N + n)*A + a
  const int b = seq / (NN * AA);
  const int na = seq % (NN * AA);
  const size_t base = ((size_t)b * HH * NN * AA + na) * DD;
  const size_t tstride = (size_t)NN * AA * DD;

  for (int i = tid; i < 48 * 64; i += 256) {
    int t = i >> 6, c = i & 63;
    sX[t * XL + c] = Hcur[base + (size_t)t * tstride + c];
  }
  __syncthreads();
  // Q,K,V = X@W + b  (3 * 3x4 tiles)
  for (int t = w; t < 36; t += 8) {
    int which = t / 12, rt = (t % 12) >> 2, ct = t & 3;
    const float* W = (which == 0) ? wq : (which == 1) ? wk : wv;
    const float* Bi = (which == 0) ? bq : (which == 1) ? bk : bv;
    float* Dst = (which == 0) ? sQ : (which == 1) ? sK : sV;
    v8f acc = bias_acc(Bi, ct * 16, lane);
    acc = wmma_acc<DD>(sX + rt * 16 * XL, XL, W + ct * 16, DD, acc, lane);
    store_tile(Dst + rt * 16 * XL + ct * 16, XL, acc, lane);
  }
  __syncthreads();
  // scores: pair p = h*3 + rowtile, each 16x48
  for (int p = w; p < 12; p += 8) {
    int h = p / 3, rt = p % 3;
    for (int ct = 0; ct < 3; ++ct) {
      v8f acc = {};
      acc = wmma_acc_bt<16>(sQ + rt * 16 * XL + h * 16, XL,
                            sK + ct * 16 * XL + h * 16, XL, acc, lane);
#pragma unroll
      for (int v = 0; v < 8; ++v) acc[v] *= 0.25f;   // 1/sqrt(16)
      store_tile(sS[p] + ct * 16, SL, acc, lane);
    }
  }
  __syncthreads();
  for (int r = tid; r < 192; r += 256) {      // softmax rows
    float* rowp = sS[r >> 4] + (r & 15) * SL;
    float mx = rowp[0];
    for (int j = 1; j < 48; ++j) mx = fmaxf(mx, rowp[j]);
    float sum = 0.f;
    for (int j = 0; j < 48; ++j) { float e = expf(rowp[j] - mx); rowp[j] = e; sum += e; }
    float inv = 1.f / sum;
    for (int j = 0; j < 48; ++j) rowp[j] *= inv;
  }
  __syncthreads();
  for (int p = w; p < 12; p += 8) {           // O = att @ V_h
    int h = p / 3, rt = p % 3;
    v8f acc = {};
    acc = wmma_acc<48>(sS[p], SL, sV + h * 16, XL, acc, lane);
    store_tile(sO + rt * 16 * XL + h * 16, XL, acc, lane);
  }
  __syncthreads();
  for (int t = w; t < 12; t += 8) {           // proj = O@wo + bo
    int rt = t >> 2, ct = t & 3;
    v8f acc = bias_acc(bo, ct * 16, lane);
    acc = wmma_acc<DD>(sO + rt * 16 * XL, XL, wo + ct * 16, DD, acc, lane);
    store_tile(sP + rt * 16 * XL + ct * 16, XL, acc, lane);
  }
  __syncthreads();
  for (int r = tid; r < 48; r += 256) {       // x1 = LN(x + proj)
    float* xr = sX + r * XL; float* pr = sP + r * XL;
    float mean = 0.f;
    for (int c = 0; c < 64; ++c) { float v = xr[c] + pr[c]; pr[c] = v; mean += v; }
    mean *= (1.f / 64.f);
    float var = 0.f;
    for (int c = 0; c < 64; ++c) { float d = pr[c] - mean; var += d * d; }
    var *= (1.f / 64.f);
    float inv = rsqrtf(var + 1e-5f);
    for (int c = 0; c < 64; ++c) xr[c] = (pr[c] - mean) * inv * ln1g[c] + ln1b[c];
  }
  __syncthreads();
  for (int t = w; t < 24; t += 8) {           // h1 = relu(x1@ff1w + ff1b)
    int rt = t >> 3, ct = t & 7;
    v8f acc = bias_acc(ff1b, ct * 16, lane);
    acc = wmma_acc<DD>(sX + rt * 16 * XL, XL, ff1w + ct * 16, DFF_, acc, lane);
    store_tile(sH1 + rt * 16 * FL + ct * 16, FL, relu8(acc), lane);
  }
  __syncthreads();
  for (int t = w; t < 12; t += 8) {           // y = h1@ff2w + ff2b
    int rt = t >> 2, ct = t & 3;
    v8f acc = bias_acc(ff2b, ct * 16, lane);
    acc = wmma_acc<DFF_>(sH1 + rt * 16 * FL, FL, ff2w + ct * 16, DD, acc, lane);
    store_tile(sP + rt * 16 * XL + ct * 16, XL, acc, lane);
  }
  __syncthreads();
  for (int r = tid; r < 48; r += 256) {       // x2 = LN(x1 + y)
    float* xr = sX + r * XL; float* pr = sP + r * XL;
    float mean = 0.f;
    for (int c = 0; c < 64; ++c) { float v = xr[c] + pr[c]; pr[c] = v; mean += v; }
    mean *= (1.f / 64.f);
    float var = 0.f;
    for (int c = 0; c < 64; ++c) { float d = pr[c] - mean; var += d * d; }
    var *= (1.f / 64.f);
    float inv = rsqrtf(var + 1e-5f);
    for (int c = 0; c < 64; ++c) xr[c] = (pr[c] - mean) * inv * ln2g[c] + ln2b[c];
  }
  __syncthreads();
  for (int i = tid; i < 48 * 64; i += 256) {
    int t = i >> 6, c = i & 63;
    Hcur[base + (size_t)t * tstride + c] = sX[t * XL + c];
  }
}

// ---------------- attribute GCN: one block per (t, chunk of 16 stations) ---
__global__ void __launch_bounds__(256) attr_kernel(
    const float* Hcur, const float* w1a, const float* w2a,
    const float* gw, const float* gb, const float* ccA,
    const float* attr_logits, float* Hsum) {
  __shared__ float sX[96 * XL];
  __shared__ float sQ[96 * XL];
  __shared__ float sK[96 * XL];
  __shared__ float sP[96 * XL];
  __shared__ float sAn[16 * 36];
  const int tid = threadIdx.x, lane = tid & 31, w = tid >> 5;
  const int t = blockIdx.x / 6;
  const int n0 = (blockIdx.x % 6) * 16;
  const size_t base = ((size_t)t * NN + n0) * AA * DD;  // 96 contiguous rows

  float l0 = attr_logits[0], l1 = attr_logits[1];
  float mx = fmaxf(l0, l1);
  float e0 = expf(l0 - mx), e1 = expf(l1 - mx);
  float wa0 = e0 / (e0 + e1), wa1 = e1 / (e0 + e1);

  for (int i = tid; i < 96 * 64; i += 256)
    sX[(i >> 6) * XL + (i & 63)] = Hcur[base + i];
  __syncthreads();
  for (int tt = w; tt < 48; tt += 8) {  // q=X@w1a, k=X@w2a
    int which = tt / 24, rt = (tt % 24) >> 2, ct = tt & 3;
    const float* W = which ? w2a : w1a;
    float* Dst = which ? sK : sQ;
    v8f acc = {};
    acc = wmma_acc<DD>(sX + rt * 16 * XL, XL, W + ct * 16, DD, acc, lane);
    store_tile(Dst + rt * 16 * XL + ct * 16, XL, acc, lane);
  }
  __syncthreads();
  for (int i = tid; i < 16 * 36; i += 256) {  // Ah = wa0*relu(qk^T)+wa1*cc + I
    int s = i / 36, r = i % 36, ii = r / 6, jj = r % 6;
    const float* qi = sQ + (s * 6 + ii) * XL;
    const float* kj = sK + (s * 6 + jj) * XL;
    float d = 0.f;
    for (int c = 0; c < 64; ++c) d += qi[c] * kj[c];
    sAn[i] = wa0 * fmaxf(d, 0.f) + wa1 * ccA[(size_t)(n0 + s) * 36 + r] +
             ((ii == jj) ? 1.f : 0.f);
  }
  __syncthreads();
  if (tid < 16) {                              // sym-normalize per station
    float* M = sAn + tid * 36;
    float dis[6];
    for (int ii = 0; ii < 6; ++ii) {
      float s = 0.f;
      for (int jj = 0; jj < 6; ++jj) s += M[ii * 6 + jj];
      dis[ii] = rsqrtf(fmaxf(s, 1e-6f));
    }
    for (int ii = 0; ii < 6; ++ii)
      for (int jj = 0; jj < 6; ++jj) M[ii * 6 + jj] *= dis[ii] * dis[jj];
  }
  __syncthreads();
  for (int hop = 0; hop < 2; ++hop) {
    for (int i = tid; i < 96 * 64; i += 256) { // P = An @ X (6x6 per station)
      int r = i >> 6, c = i & 63, s = r / 6, ii = r % 6;
      const float* M = sAn + s * 36 + ii * 6;
      const float* Xs = sX + s * 6 * XL + c;
      float acc = 0.f;
#pragma unroll
      for (int jj = 0; jj < 6; ++jj) acc += M[jj] * Xs[jj * XL];
      sP[r * XL + c] = acc;
    }
    __syncthreads();
    for (int tt = w; tt < 24; tt += 8) {       // X = relu(P@gw + gb)
      int rt = tt >> 2, ct = tt & 3;
      v8f acc = bias_acc(gb, ct * 16, lane);
      acc = wmma_acc<DD>(sP + rt * 16 * XL, XL, gw + ct * 16, DD, acc, lane);
      store_tile(sX + rt * 16 * XL + ct * 16, XL, relu8(acc), lane);
    }
    __syncthreads();
  }
  for (int i = tid; i < 96 * 64; i += 256)
    Hsum[base + i] = sX[(i >> 6) * XL + (i & 63)];
}

// ---------------- spatial GCN: one block per (t, a) ------------------------
__global__ void __launch_bounds__(256) spa_kernel(
    const float* Hcur, const float* w1s, const float* w2s,
    const float* gw, const float* gb, const float* ccS, const float* A_geo,
    const float* spa_logits, float* Hsum) {
  __shared__ float sX[96 * XL];
  __shared__ float sQ[96 * XL];
  __shared__ float sK[96 * XL];
  __shared__ float sP[96 * XL];
  __shared__ float sS[96 * SLD];
  __shared__ float sDis[96];
  const int tid = threadIdx.x, lane = tid & 31, w = tid >> 5;
  const int t = blockIdx.x / 6;
  const int a = blockIdx.x % 6;
  const size_t base = ((size_t)t * NN * AA + a) * DD;
  const size_t rstride = (size_t)AA * DD;

  float l0 = spa_logits[0], l1 = spa_logits[1], l2 = spa_logits[2];
  float mx = fmaxf(fmaxf(l0, l1), l2);
  float e0 = expf(l0 - mx), e1 = expf(l1 - mx), e2 = expf(l2 - mx);
  float inv = 1.f / (e0 + e1 + e2);
  float ws0 = e0 * inv, ws1 = e1 * inv, ws2 = e2 * inv;

  for (int i = tid; i < 96 * 64; i += 256) {
    int r = i >> 6, c = i & 63;
    sX[r * XL + c] = Hcur[base + (size_t)r * rstride + c];
  }
  __syncthreads();
  for (int tt = w; tt < 48; tt += 8) {  // qs, ks
    int which = tt / 24, rt = (tt % 24) >> 2, ct = tt & 3;
    const float* W = which ? w2s : w1s;
    float* Dst = which ? sK : sQ;
    v8f acc = {};
    acc = wmma_acc<DD>(sX + rt * 16 * XL, XL, W + ct * 16, DD, acc, lane);
    store_tile(Dst + rt * 16 * XL + ct * 16, XL, acc, lane);
  }
  __syncthreads();
  for (int tt = w; tt < 36; tt += 8) {  // Ah = ws0*relu(qs ks^T)+ws1*cc+ws2*geo+I
    int rt = tt / 6, ct = tt % 6;
    v8f acc = {};
    acc = wmma_acc_bt<DD>(sQ + rt * 16 * XL, XL, sK + ct * 16 * XL, XL, acc, lane);
    const int n = lane & 15, sel = lane >> 4;
#pragma unroll
    for (int v = 0; v < 8; ++v) {
      int ri = rt * 16 + v + 8 * sel;
      int ci = ct * 16 + n;
      sS[ri * SLD + ci] = ws0 * fmaxf(acc[v], 0.f) +
                          ws1 * ccS[((size_t)a * NN + ri) * NN + ci] +
                          ws2 * A_geo[ri * NN + ci] + ((ri == ci) ? 1.f : 0.f);
    }
  }
  __syncthreads();
  for (int r = tid; r < 96; r += 256) {
    float s = 0.f;
    const float* rp = sS + r * SLD;
    for (int j = 0; j < 96; ++j) s += rp[j];
    sDis[r] = rsqrtf(fmaxf(s, 1e-6f));
  }
  __syncthreads();
  for (int i = tid; i < 96 * 96; i += 256) {
    int r = i / 96, c = i % 96;
    sS[r * SLD + c] *= sDis[r] * sDis[c];
  }
  __syncthreads();
  for (int hop = 0; hop < 2; ++hop) {
    for (int tt = w; tt < 24; tt += 8) {       // P = An @ X (K=96)
      int rt = tt >> 2, ct = tt & 3;
      v8f acc = {};
      acc = wmma_acc<96>(sS + rt * 16 * SLD, SLD, sX + ct * 16, XL, acc, lane);
      store_tile(sP + rt * 16 * XL + ct * 16, XL, acc, lane);
    }
    __syncthreads();
    for (int tt = w; tt < 24; tt += 8) {       // X = relu(P@gw + gb)
      int rt = tt >> 2, ct = tt & 3;
      v8f acc = bias_acc(gb, ct * 16, lane);
      acc = wmma_acc<DD>(sP + rt * 16 * XL, XL, gw + ct * 16, DD, acc, lane);
      store_tile(sX + rt * 16 * XL + ct * 16, XL, relu8(acc), lane);
    }
    __syncthreads();
  }
  for (int i = tid; i < 96 * 64; i += 256) {   // accumulate onto Hattr
    int r = i >> 6, c = i & 63;
    size_t g = base + (size_t)r * rstride + c;
    Hsum[g] += sX[r * XL + c];
  }
}

// ---------------- fuse MLP: (Hattr+Hspa) 64 -> 128 -> 64 -------------------
__global__ void __launch_bounds__(256) fuse_kernel(
    const float* Hsum, const float* w1, const float* b1, const float* w2,
    const float* b2, float* Hcur) {
  __shared__ float sX[64 * XL];
  __shared__ float sH1[64 * FL];
  const int tid = threadIdx.x, lane = tid & 31, w = tid >> 5;
  const size_t row0 = (size_t)blockIdx.x * 64;
  for (int i = tid; i < 64 * 64; i += 256)
    sX[(i >> 6) * XL + (i & 63)] = Hsum[row0 * DD + i];
  __syncthreads();
  for (int t = w; t < 32; t += 8) {
    int rt = t >> 3, ct = t & 7;
    v8f acc = bias_acc(b1, ct * 16, lane);
    acc = wmma_acc<DD>(sX + rt * 16 * XL, XL, w1 + ct * 16, HID_, acc, lane);
    store_tile(sH1 + rt * 16 * FL + ct * 16, FL, relu8(acc), lane);
  }
  __syncthreads();
  for (int t = w; t < 16; t += 8) {
    int rt = t >> 2, ct = t & 3;
    v8f acc = bias_acc(b2, ct * 16, lane);
    acc = wmma_acc<HID_>(sH1 + rt * 16 * FL, FL, w2 + ct * 16, DD, acc, lane);
    const int n = lane & 15, sel = lane >> 4;
#pragma unroll
    for (int v = 0; v < 8; ++v)
      Hcur[(row0 + rt * 16 + v + 8 * sel) * DD + ct * 16 + n] = acc[v];
  }
}

extern "C" void kernel_launch(void* const* d_in, const int* in_sizes, int n_in,
                              void* d_out, int out_size, void* d_ws,
                              size_t ws_size, hipStream_t stream) {
  (void)in_sizes; (void)n_in; (void)out_size; (void)ws_size;
  const float* X_in = (const float*)d_in[0];
  const float* M_in = (const float*)d_in[1];
  const float* U_in = (const float*)d_in[2];
  const float* W_in = (const float*)d_in[3];
  const float* A_geo = (const float*)d_in[4];
  const float* c_attr = (const float*)d_in[5];
  const float* emb_w1 = (const float*)d_in[6];
  const float* emb_b1 = (const float*)d_in[7];
  const float* emb_w2 = (const float*)d_in[8];
  const float* emb_b2 = (const float*)d_in[9];
  const float* wq = (const float*)d_in[10];
  const float* bq = (const float*)d_in[11];
  const float* wk = (const float*)d_in[12];
  const float* bk = (const float*)d_in[13];
  const float* wv = (const float*)d_in[14];
  const float* bv = (const float*)d_in[15];
  const float* wo = (const float*)d_in[16];
  const float* bo = (const float*)d_in[17];
  const float* ln1g = (const float*)d_in[18];
  const float* ln1b = (const float*)d_in[19];
  const float* ln2g = (const float*)d_in[20];
  const float* ln2b = (const float*)d_in[21];
  const float* ff1w = (const float*)d_in[22];
  const float* ff1b = (const float*)d_in[23];
  const float* ff2w = (const float*)d_in[24];
  const float* ff2b = (const float*)d_in[25];
  const float* w1_attr = (const float*)d_in[26];
  const float* w2_attr = (const float*)d_in[27];
  const float* w1_spa = (const float*)d_in[28];
  const float* w2_spa = (const float*)d_in[29];
  const float* gcn_attr_w = (const float*)d_in[30];
  const float* gcn_attr_b = (const float*)d_in[31];
  const float* gcn_spa_w = (const float*)d_in[32];
  const float* gcn_spa_b = (const float*)d_in[33];
  const float* attr_logits = (const float*)d_in[34];
  const float* spa_logits = (const float*)d_in[35];
  const float* fuse_w1 = (const float*)d_in[36];
  const float* fuse_b1 = (const float*)d_in[37];
  const float* fuse_w2 = (const float*)d_in[38];
  const float* fuse_b2 = (const float*)d_in[39];

  float* Hcur = (float*)d_out;
  const size_t TOT = (size_t)ROWS * DD;
  float* Hsum = (float*)d_ws;          // Hattr, then spa accumulates
  float* ccA = Hsum + TOT;
  float* ccS = ccA + (size_t)NN * 36;

  cc_attr_kernel<<<(NN * 36 + 255) / 256, 256, 0, stream>>>(c_attr, ccA);
  cc_spa_kernel<<<(AA * NN * NN + 255) / 256, 256, 0, stream>>>(c_attr, ccS);
  embed_kernel<<<ROWS / 64, 256, 0, stream>>>(X_in, M_in, U_in, W_in, c_attr,
                                              emb_w1, emb_b1, emb_w2, emb_b2,
                                              Hcur);
  for (int sb = 0; sb < 3; ++sb) {
    temporal_kernel<<<BB * NN * AA, 256, 0, stream>>>(
        Hcur, wq, bq, wk, bk, wv, bv, wo, bo, ln1g, ln1b, ln2g, ln2b, ff1w,
        ff1b, ff2w, ff2b);
    attr_kernel<<<BHDIM * 6, 256, 0, stream>>>(Hcur, w1_attr, w2_attr,
                                               gcn_attr_w, gcn_attr_b, ccA,
                                               attr_logits, Hsum);
    spa_kernel<<<BHDIM * 6, 256, 0, stream>>>(Hcur, w1_spa, w2_spa, gcn_spa_w,
                                              gcn_spa_b, ccS, A_geo,
                                              spa_logits, Hsum);
    fuse_kernel<<<ROWS / 64, 256, 0, stream>>>(Hsum, fuse_w1, fuse_b1, fuse_w2,
                                               fuse_b2, Hcur);
  }
}